// TwistorLNN_37572373905543
// MI455X (gfx1250) — compile-verified
//
#include <hip/hip_runtime.h>

typedef float v2f __attribute__((ext_vector_type(2)));
typedef float v8f __attribute__((ext_vector_type(8)));

#define T_STEPS 4096
#define BATCH   256
#define IN      64
#define HID     16
#define DT_C    0.1f

__device__ __forceinline__ v8f wmma4(v2f a, v2f b, v8f c) {
    // D = A(16x4) x B(4x16) + C(16x16), all f32
    return __builtin_amdgcn_wmma_f32_16x16x4_f32(
        /*neg_a=*/false, a, /*neg_b=*/false, b,
        /*c_mod=*/(short)0, c, /*reuse_a=*/false, /*reuse_b=*/false);
}

// Cross half-wave swap (lane ^ 16) as pure VALU: v_permlanex16_b32 with
// identity lane selects. No DS pipeline, no dscnt wait.
__device__ __forceinline__ float xhalf(float v) {
    int i = __float_as_int(v);
    int r = __builtin_amdgcn_permlanex16(i, i, 0x76543210, 0xFEDCBA98,
                                         false, false);
    return __int_as_float(r);
}

// Fast hardware approximations (v_rcp_f32 / v_sqrt_f32) -- single TRANS ops
// that co-execute with WMMA, instead of the correctly-rounded
// div_scale/div_fmas/div_fixup expansions __frcp_rn/__fsqrt_rn produce.
__device__ __forceinline__ float frcp(float v)  { return __builtin_amdgcn_rcpf(v); }
__device__ __forceinline__ float fsqrt(float v) { return __builtin_amdgcn_sqrtf(v); }

// One wave handles 16 batch elements for the whole T=4096 scan.
// State layout (transposed): z[h, b] in WMMA C-layout:
//   VGPR j, lanes 0-15 : (h = j,   b = lane)
//   VGPR j, lanes 16-31: (h = j+8, b = lane-16)
__global__ __launch_bounds__(32)
void twistor_lnn_scan(const float* __restrict__ x,
                      const float* __restrict__ Wr,
                      const float* __restrict__ Wi,
                      const float* __restrict__ Ur,
                      const float* __restrict__ Ui,
                      const float* __restrict__ out_w,
                      const float* __restrict__ out_b,
                      float* __restrict__ y)
{
    const int lane = threadIdx.x & 31;
    const int row  = lane & 15;   // M row (weights) / batch col (B,C frags)
    const int hi   = lane >> 4;   // which half-wave
    const int b0   = blockIdx.x * 16;

    // ---- Preload loop-invariant A-fragments (16x4 f32 layout):
    //   frag.x = M[row][kb + 2*hi], frag.y = M[row][kb + 2*hi + 1]
    v2f WrA[4], WiA[4], WiN[4];
#pragma unroll
    for (int c = 0; c < 4; ++c) {
        const int k = 4 * c + 2 * hi;
        WrA[c].x = Wr[row * HID + k];  WrA[c].y = Wr[row * HID + k + 1];
        WiA[c].x = Wi[row * HID + k];  WiA[c].y = Wi[row * HID + k + 1];
        WiN[c].x = -WiA[c].x;          WiN[c].y = -WiA[c].y;   // for  -Wi*ti
    }
    v2f UrA[16], UiA[16];
#pragma unroll
    for (int c = 0; c < 16; ++c) {
        const int k = 4 * c + 2 * hi;
        UrA[c].x = Ur[row * IN + k];   UrA[c].y = Ur[row * IN + k + 1];
        UiA[c].x = Ui[row * IN + k];   UiA[c].y = Ui[row * IN + k + 1];
    }
    float ow[8];
#pragma unroll
    for (int j = 0; j < 8; ++j) ow[j] = out_w[j + 8 * hi];
    const float ob = out_b[0];

    v8f zr = {}; v8f zi = {};   // z0 = 0

    const float* xrow = x + (size_t)(b0 + row) * IN;

    for (int t = 0; t < T_STEPS; ++t) {
        const float* xt = xrow + (size_t)t * (BATCH * IN);

        // ---- x tile directly in B-fragment layout (K=i, N=batch):
        //   VGPR0 holds K=4c (lo lanes) / 4c+2 (hi lanes) -> .x; VGPR1 -> .y
        v2f xb[16];
#pragma unroll
        for (int c = 0; c < 16; ++c)
            xb[c] = *reinterpret_cast<const v2f*>(xt + 4 * c + 2 * hi);

        if (t + 4 < T_STEPS)
            __builtin_prefetch(xt + 4 * (BATCH * IN), 0, 0);

        // ---- complex tanh, elementwise on state layout.
        // tanh(a+ib) = (sinh2a + i sin2b)/(cosh2a + cos2b); multiplying
        // num/den by 2e (e = exp(2a)) removes the 1/e reciprocal:
        //   = ((e^2-1) + i 2e sin2b) / ((e^2+1) + 2e cos2b)
        float tr[8], ti[8];
#pragma unroll
        for (int j = 0; j < 8; ++j) {
            float a = zr[j], b = zi[j];
            float ac  = fminf(fmaxf(a, -15.f), 15.f);
            float e   = __expf(2.f * ac);
            float e2  = e * e;
            float s2b = __sinf(2.f * b), c2b = __cosf(2.f * b);
            float den = fmaf(2.f * e, c2b, e2 + 1.f);
            float r   = frcp(den);
            tr[j] = (e2 - 1.f) * r;
            ti[j] = (2.f * e) * s2b * r;
        }

        // ---- C-layout -> B-fragment layout for tanh values.
        // Lane needs (K = 4c + 2*hi + {0,1}, b = row); it lives locally iff
        // (c>=2) == (hi==1), otherwise in the partner lane (lane^16).
        v2f trB[4], tiB[4];
#pragma unroll
        for (int c = 0; c < 4; ++c) {
            const int mA = (4 * c) & 7;       // reg index hi==0 lanes need
            const int mB = (4 * c + 2) & 7;   // reg index hi==1 lanes need
            const bool local = ((c >= 2) == (hi == 1));
            // payload = value the PARTNER lane needs from us
            float rr0 = xhalf(hi ? tr[mA]     : tr[mB]);
            float rr1 = xhalf(hi ? tr[mA + 1] : tr[mB + 1]);
            float ri0 = xhalf(hi ? ti[mA]     : ti[mB]);
            float ri1 = xhalf(hi ? ti[mA + 1] : ti[mB + 1]);
            float lr0 = hi ? tr[mB]     : tr[mA];
            float lr1 = hi ? tr[mB + 1] : tr[mA + 1];
            float li0 = hi ? ti[mB]     : ti[mA];
            float li1 = hi ? ti[mB + 1] : ti[mA + 1];
            trB[c].x = local ? lr0 : rr0;  trB[c].y = local ? lr1 : rr1;
            tiB[c].x = local ? li0 : ri0;  tiB[c].y = local ? li1 : ri1;
        }

        // ---- 4 balanced WMMA accumulator chains, 12 deep each:
        //   R0 = recurrent-real(8) + Ur*x chunks 0..3   (12)
        //   R1 = Ur*x chunks 4..15                      (12)
        //   I0 = recurrent-imag(8) + Ui*x chunks 0..3   (12)
        //   I1 = Ui*x chunks 4..15                      (12)
        v8f aR0 = {}, aR1 = {}, aI0 = {}, aI1 = {};
#pragma unroll
        for (int c = 0; c < 4; ++c) {
            aR0 = wmma4(WrA[c], trB[c], aR0);   // + Wr * tr
            aR0 = wmma4(WiN[c], tiB[c], aR0);   // - Wi * ti
            aI0 = wmma4(WrA[c], tiB[c], aI0);   // + Wr * ti
            aI0 = wmma4(WiA[c], trB[c], aI0);   // + Wi * tr
        }
#pragma unroll
        for (int c = 0; c < 4; ++c) {
            aR0 = wmma4(UrA[c], xb[c], aR0);
            aI0 = wmma4(UiA[c], xb[c], aI0);
        }
#pragma unroll
        for (int c = 4; c < 16; ++c) {
            aR1 = wmma4(UrA[c], xb[c], aR1);
            aI1 = wmma4(UiA[c], xb[c], aI1);
        }

        // ---- Euler update with tau = 1 + |z|, plus readout partials
        float part = 0.f;
#pragma unroll
        for (int j = 0; j < 8; ++j) {
            float a   = zr[j], b = zi[j];
            float it  = frcp(1.f + fsqrt(fmaf(a, a, b * b)));
            float nr  = aR0[j] + aR1[j] - a;
            float ni  = aI0[j] + aI1[j] - b;
            a += DT_C * nr * it;
            b += DT_C * ni * it;
            zr[j] = a; zi[j] = b;
            part = fmaf(a, ow[j], part);        // real part * out_w
        }
        float part2 = xhalf(part);
        if (hi == 0)
            y[(size_t)t * BATCH + b0 + row] = part + part2 + ob;
    }
}

extern "C" void kernel_launch(void* const* d_in, const int* in_sizes, int n_in,
                              void* d_out, int out_size, void* d_ws, size_t ws_size,
                              hipStream_t stream) {
    const float* x    = (const float*)d_in[0];
    const float* Wr   = (const float*)d_in[1];
    const float* Wi   = (const float*)d_in[2];
    const float* Ur   = (const float*)d_in[3];
    const float* Ui   = (const float*)d_in[4];
    const float* outw = (const float*)d_in[5];
    const float* outb = (const float*)d_in[6];
    float* y = (float*)d_out;

    // 256 batches / 16 per wave -> 16 single-wave workgroups, each runs the
    // full sequential scan independently (no inter-wave sync needed).
    twistor_lnn_scan<<<dim3(BATCH / 16), dim3(32), 0, stream>>>(
        x, Wr, Wi, Ur, Ui, outw, outb, y);
}